// DotAttention_19146964205783
// MI455X (gfx1250) — compile-verified
//
#include <hip/hip_runtime.h>
#include <hip/hip_bf16.h>

// DotAttention, B=32, S=8192, D=1024, fp32.
// Memory-bound: keys (1 GiB) + values (1 GiB) streamed once @ 23.3 TB/s ~= 90 us.
// Scores computed with V_WMMA_F32_16X16X4_F32 (full fp32 precision).

#define BB   32
#define SS   8192
#define DD   1024
#define WPB  8                 // waves per block in score kernel (wave32)
#define SSPLIT 64              // S partitions for context pass
#define SCHUNK (SS / SSPLIT)   // 128 rows per context block

typedef float v2f __attribute__((ext_vector_type(2)));
typedef float v4f __attribute__((ext_vector_type(4)));   // clang ext-vector: OK for NT builtins
typedef float v8f __attribute__((ext_vector_type(8)));

// monotonic float <-> signed-int key (exact, order-independent atomic max)
__device__ __forceinline__ int   fkey(float f)  { int i = __float_as_int(f); return i >= 0 ? i : (i ^ 0x7FFFFFFF); }
__device__ __forceinline__ float funkey(int i)  { return __int_as_float(i >= 0 ? i : (i ^ 0x7FFFFFFF)); }

__global__ void k_init_max(int* __restrict__ maxkey) {
    if (threadIdx.x < BB) maxkey[threadIdx.x] = (int)0x80000000;
}

// ---------------- scores[b,s] = q[b,:] . K[b,s,:]  via fp32 WMMA -----------------
// Per wave: 16 key rows. A(16x4)=keys slice (documented 32-bit A layout: lane L
// holds row L%16; VGPR0/1 = K pair selected by L/16 -> one global_load_b64).
// B(4x16)=query slice broadcast over N (ds_load_b64, same addr within lane half).
// C(16x16) = scores replicated over N; lanes 0 and 16 own the 16 results.
__global__ void __launch_bounds__(256) k_scores(const float* __restrict__ q,
                                                const float* __restrict__ keys,
                                                float* __restrict__ scores,
                                                int* __restrict__ maxkey) {
    __shared__ float qs[DD];            // staged query row (4 KB)
    __shared__ float sstage[WPB * 16];  // 128 scores staged for coalesced store
    __shared__ float red[256];
    const int b   = blockIdx.y;
    const int tid = threadIdx.x;

    ((v4f*)qs)[tid] = ((const v4f*)(q + (size_t)b * DD))[tid];
    __syncthreads();

    const int wave = tid >> 5, lane = tid & 31;
    const int row  = lane & 15;           // M index
    const int col0 = (lane >> 4) * 2;     // K pair: lanes 0-15 -> K0/K1, 16-31 -> K2/K3
    const int blk_s0 = blockIdx.x * (WPB * 16);
    const int s0     = blk_s0 + wave * 16;

    const float* kbase = keys + ((size_t)b * SS + (size_t)(s0 + row)) * DD + col0;
    const float* qbase = qs + col0;

    v8f c = {0.f, 0.f, 0.f, 0.f, 0.f, 0.f, 0.f, 0.f};
    #pragma unroll 8                       // 8 b64 loads in flight per wave
    for (int d0 = 0; d0 < DD; d0 += 4) {
        v2f a  = *(const v2f*)(kbase + d0);   // global_load_b64 (keys stream, RT: intra-wave line reuse)
        v2f bv = *(const v2f*)(qbase + d0);   // ds_load_b64 broadcast
        c = __builtin_amdgcn_wmma_f32_16x16x4_f32(false, a, false, bv,
                                                  (short)0, c, false, false);
    }

    // C layout: VGPR r, lanes 0-15 -> M=r, lanes 16-31 -> M=r+8 (replicated over N).
    // Stage 16 scores/wave in LDS, then one coalesced 128-float store per block.
    if (lane == 0) {
        #pragma unroll
        for (int i = 0; i < 8; ++i) sstage[wave * 16 + i] = c[i];
    }
    if (lane == 16) {
        #pragma unroll
        for (int i = 0; i < 8; ++i) sstage[wave * 16 + 8 + i] = c[i];
    }

    // block max -> global ordered-int atomic max (exact => deterministic)
    float m = c[0];
    #pragma unroll
    for (int i = 1; i < 8; ++i) m = fmaxf(m, c[i]);
    red[tid] = m;
    __syncthreads();

    if (tid < WPB * 16)
        scores[(size_t)b * SS + blk_s0 + tid] = sstage[tid];

    for (int st = 128; st > 0; st >>= 1) {
        if (tid < st) red[tid] = fmaxf(red[tid], red[tid + st]);
        __syncthreads();
    }
    if (tid == 0) atomicMax(maxkey + b, fkey(red[0]));
}

// ---------------- per-batch sum(exp(s - max)) -> 1/sum (deterministic tree) ------
__global__ void __launch_bounds__(256) k_stats(const float* __restrict__ scores,
                                               const int* __restrict__ maxkey,
                                               float* __restrict__ stat_max,
                                               float* __restrict__ stat_inv) {
    __shared__ float red[256];
    const int b = blockIdx.x, tid = threadIdx.x;
    const float mx = funkey(maxkey[b]);
    const float* s = scores + (size_t)b * SS;
    float sum = 0.f;
    for (int j = tid; j < SS; j += 256) sum += expf(s[j] - mx);   // fixed order
    red[tid] = sum;
    __syncthreads();
    for (int st = 128; st > 0; st >>= 1) {
        if (tid < st) red[tid] += red[tid + st];
        __syncthreads();
    }
    if (tid == 0) { stat_max[b] = mx; stat_inv[b] = 1.0f / red[0]; }
}

// ---------------- weights + partial context (streams V once, NT b128 loads) ------
__global__ void __launch_bounds__(256) k_ctx(const float* __restrict__ scores,
                                             const float* __restrict__ values,
                                             const float* __restrict__ stat_max,
                                             const float* __restrict__ stat_inv,
                                             float* __restrict__ weights_out,
                                             float* __restrict__ partials) {
    __shared__ float wl[SCHUNK];
    const int split = blockIdx.x, b = blockIdx.y, tid = threadIdx.x;
    const int s0 = split * SCHUNK;
    const float mx = stat_max[b], inv = stat_inv[b];

    if (tid < SCHUNK) {
        float w = expf(scores[(size_t)b * SS + s0 + tid] - mx) * inv;
        wl[tid] = w;
        weights_out[(size_t)b * SS + s0 + tid] = w;   // each (b,s) written exactly once
    }
    __syncthreads();

    const v4f* v4 = (const v4f*)(values + ((size_t)b * SS + s0) * DD) + tid;
    v4f acc = {0.f, 0.f, 0.f, 0.f};
    #pragma unroll 4
    for (int j = 0; j < SCHUNK; ++j) {                // fixed order => deterministic
        const float w = wl[j];
        // values are strictly use-once per cacheline -> non-temporal, keep L2 clean
        const v4f v = __builtin_nontemporal_load(v4 + (size_t)j * (DD / 4));
        acc += w * v;
    }
    ((v4f*)partials)[((size_t)b * SSPLIT + split) * (DD / 4) + tid] = acc;
}

// ---------------- reduce SSPLIT partials -> context ------------------------------
__global__ void __launch_bounds__(256) k_reduce(const float* __restrict__ partials,
                                                float* __restrict__ ctx) {
    const int idx = blockIdx.x * 256 + threadIdx.x;   // over B*D/4 = 8192
    const int b  = idx / (DD / 4);
    const int d4 = idx % (DD / 4);
    const v4f* p = (const v4f*)partials + (size_t)b * SSPLIT * (DD / 4) + d4;
    v4f acc = {0.f, 0.f, 0.f, 0.f};
    for (int s = 0; s < SSPLIT; ++s)
        acc += p[(size_t)s * (DD / 4)];
    ((v4f*)ctx)[idx] = acc;
}

extern "C" void kernel_launch(void* const* d_in, const int* in_sizes, int n_in,
                              void* d_out, int out_size, void* d_ws, size_t ws_size,
                              hipStream_t stream) {
    const float* q    = (const float*)d_in[0];   // [B, D]
    const float* keys = (const float*)d_in[1];   // [B, S, D]
    const float* vals = (const float*)d_in[2];   // [B, S, D]

    float* out_ctx = (float*)d_out;                    // [B, D]   (tuple elem 0)
    float* out_w   = (float*)d_out + (size_t)BB * DD;  // [B, S]   (tuple elem 1)

    // workspace layout (floats): scores | maxkey | stat_max | stat_inv | partials
    float* ws       = (float*)d_ws;
    float* scores   = ws;                                    // B*S = 262144 floats
    int*   maxkey   = (int*)(ws + (size_t)BB * SS);          // B ints
    float* stat_max = ws + (size_t)BB * SS + 64;             // B floats
    float* stat_inv = ws + (size_t)BB * SS + 128;            // B floats
    float* partials = ws + (size_t)BB * SS + 256;            // B*SSPLIT*D floats (8 MB)

    k_init_max<<<1, 32, 0, stream>>>(maxkey);
    k_scores<<<dim3(SS / (16 * WPB), BB), 256, 0, stream>>>(q, keys, scores, maxkey);
    k_stats<<<BB, 256, 0, stream>>>(scores, maxkey, stat_max, stat_inv);
    k_ctx<<<dim3(SSPLIT, BB), 256, 0, stream>>>(scores, vals, stat_max, stat_inv,
                                                out_w, partials);
    k_reduce<<<(BB * DD / 4) / 256, 256, 0, stream>>>(partials, out_ctx);
}